// Grover2UnimolEmbedding_63007170232457
// MI455X (gfx1250) — compile-verified
//
#include <hip/hip_runtime.h>
#include <hip/hip_bf16.h>

typedef __attribute__((ext_vector_type(2))) float v2f;
typedef __attribute__((ext_vector_type(8))) float v8f;

#define B_MOLS   128
#define NHEAD    16
#define D_MODEL  512
#define D_IN     256
#define N_ATOMS  128
#define K_TOT    512          // 2 * D_IN
#define LDS_AK   516          // padded row stride (floats) to avoid bank conflicts

// -------------------------------------------------------------------------
// Kernel 1: atoms = concat(f_atoms, f_atoms_out) @ W_atom + b_atom,
// fused with the padded gather + transpose into atoms_emb[p][mol][n].
// grid = (4 ngroups of 128 cols, 8 p-tiles of 16 rows, 128 mols), 256 thr.
// -------------------------------------------------------------------------
__global__ __launch_bounds__(256)
void atoms_emb_kernel(const float* __restrict__ f_atoms,
                      const float* __restrict__ f_atoms_out,
                      const int*   __restrict__ a_scope,   // [B][2] = start,len
                      const float* __restrict__ W_atom,    // [512][512]
                      const float* __restrict__ b_atom,    // [512]
                      float*       __restrict__ out1)      // [128][128][512]
{
    __shared__ float lds_a[16 * LDS_AK];   // 16 rows x 512 K (padded)

    const int mol    = blockIdx.z;
    const int ptile  = blockIdx.y;
    const int ngroup = blockIdx.x;
    const int start  = a_scope[2 * mol];
    const int len    = a_scope[2 * mol + 1];
    const int tid    = threadIdx.x;

    // Cooperative load: 16 rows x 512 floats = 2048 float4 slots / 256 thr.
    #pragma unroll
    for (int i = 0; i < 8; ++i) {
        int slot = tid + i * 256;          // 0..2047
        int row  = slot >> 7;              // /128
        int q    = slot & 127;
        int k    = q << 2;
        int p    = ptile * 16 + row;
        float4 v = make_float4(0.f, 0.f, 0.f, 0.f);
        if (p < len) {
            long g = (long)(start + p);
            const float* src = (k < D_IN) ? (f_atoms + g * D_IN + k)
                                          : (f_atoms_out + g * D_IN + (k - D_IN));
            v = *reinterpret_cast<const float4*>(src);
        }
        *reinterpret_cast<float4*>(&lds_a[row * LDS_AK + k]) = v;
    }
    __syncthreads();

    const int wave = tid >> 5;     // 0..7  -> one 16-col tile each
    const int lane = tid & 31;
    const int lg   = lane >> 4;    // lane half: selects K pair / M+8
    const int ln   = lane & 15;    // A: M row, B/C/D: N col
    const int n0   = ngroup * 128 + wave * 16;

    v8f acc = {};
    for (int k0 = 0; k0 < K_TOT; k0 += 4) {
        const int ka = k0 + 2 * lg;
        // A 16x4 f32 layout: lane%16 = M, VGPR0/1 = K = 2*(lane/16)+{0,1}
        v2f a = *reinterpret_cast<const v2f*>(&lds_a[ln * LDS_AK + ka]);
        // B 4x16 f32 layout: lane%16 = N, VGPR0/1 = K = 2*(lane/16)+{0,1}
        v2f b;
        b.x = W_atom[(size_t)ka * D_MODEL + n0 + ln];
        b.y = W_atom[(size_t)(ka + 1) * D_MODEL + n0 + ln];
        acc = __builtin_amdgcn_wmma_f32_16x16x4_f32(
            false, a, false, b, (short)0, acc, false, false);
    }

    const float bias = b_atom[n0 + ln];
    #pragma unroll
    for (int v = 0; v < 8; ++v) {
        const int M = v + 8 * lg;          // C/D: VGPR v -> M, lane half -> +8
        const int p = ptile * 16 + M;
        const float val = (p < len) ? (acc[v] + bias) : 0.0f;
        out1[((size_t)p * B_MOLS + mol) * D_MODEL + n0 + ln] = val;
    }
}

// -------------------------------------------------------------------------
// Kernel 2: bonds = concat(f_bonds, f_bonds_out) @ W_bond + b_bond,
// fused with the scatter into apairs[mol][h][r][c].
// One wave per 16-bond tile, 8 tiles per 256-thread block.
// -------------------------------------------------------------------------
__global__ __launch_bounds__(256)
void bonds_scatter_kernel(const float* __restrict__ f_bonds,
                          const float* __restrict__ f_bonds_out,
                          const int*   __restrict__ b2a,
                          const int*   __restrict__ b2revb,
                          const int*   __restrict__ b_scope,  // [B][2]
                          const float* __restrict__ W_bond,   // [512][16]
                          const float* __restrict__ b_bond,   // [16]
                          float*       __restrict__ out2,     // [B][16][128][128]
                          int Nb)
{
    __shared__ float lds_w[K_TOT * NHEAD];     // 32 KB
    __shared__ int   lds_starts[B_MOLS];

    const int tid = threadIdx.x;
    #pragma unroll
    for (int i = 0; i < 8; ++i) {
        int slot = tid + i * 256;              // 2048 float4 slots
        *reinterpret_cast<float4*>(&lds_w[slot * 4]) =
            *reinterpret_cast<const float4*>(&W_bond[slot * 4]);
    }
    if (tid < B_MOLS) lds_starts[tid] = b_scope[2 * tid];
    __syncthreads();

    const int wave = tid >> 5;
    const int lane = tid & 31;
    const int lg   = lane >> 4;
    const int ln   = lane & 15;
    const int b0   = (blockIdx.x * 8 + wave) * 16;
    if (b0 >= Nb) return;                      // uniform per wave

    const int  row    = b0 + ln;
    const bool rvalid = row < Nb;
    const float* rf0  = f_bonds     + (size_t)row * D_IN;
    const float* rf1  = f_bonds_out + (size_t)row * D_IN;

    v8f acc = {};
    for (int k0 = 0; k0 < K_TOT; k0 += 4) {
        const int ka = k0 + 2 * lg;
        v2f a = {0.f, 0.f};
        if (rvalid) {
            const float* src = (ka < D_IN) ? (rf0 + ka) : (rf1 + (ka - D_IN));
            a = *reinterpret_cast<const v2f*>(src);
        }
        v2f b;
        b.x = lds_w[ka * NHEAD + ln];
        b.y = lds_w[(ka + 1) * NHEAD + ln];
        acc = __builtin_amdgcn_wmma_f32_16x16x4_f32(
            false, a, false, b, (short)0, acc, false, false);
    }

    const float bias = b_bond[ln];             // N = head = ln
    #pragma unroll
    for (int v = 0; v < 8; ++v) {
        const int bb = b0 + v + 8 * lg;        // bond index = M row of tile
        if (bb < Nb) {
            const float val = acc[v] + bias;
            // mol = clamp(count(starts <= bb) - 1, 0)
            int lo = 0, hi = B_MOLS;
            while (lo < hi) {
                int mid = (lo + hi) >> 1;
                if (lds_starts[mid] <= bb) lo = mid + 1; else hi = mid;
            }
            int m = lo - 1; if (m < 0) m = 0;
            const int c = b2a[bb];
            const int r = b2a[b2revb[bb]];
            out2[(((size_t)m * NHEAD + ln) * N_ATOMS + r) * N_ATOMS + c] = val;
        }
    }
}

// -------------------------------------------------------------------------
// Kernel 3: initialize apairs to 0 (valid col) / -inf (padded col).
// -------------------------------------------------------------------------
__global__ __launch_bounds__(256)
void apairs_init_kernel(const int* __restrict__ a_scope,
                        float*     __restrict__ out2)
{
    const size_t idx4 = (size_t)blockIdx.x * 256 + threadIdx.x;   // float4 idx
    const size_t e    = idx4 * 4;
    const int c0  = (int)(e & (N_ATOMS - 1));
    const int m   = (int)(e >> 18);            // 16*128*128 = 2^18 per mol
    const int len = a_scope[2 * m + 1];
    const float NEG = -__builtin_inff();
    float4 v;
    v.x = (c0 + 0 < len) ? 0.f : NEG;
    v.y = (c0 + 1 < len) ? 0.f : NEG;
    v.z = (c0 + 2 < len) ? 0.f : NEG;
    v.w = (c0 + 3 < len) ? 0.f : NEG;
    *reinterpret_cast<float4*>(out2 + e) = v;
}

// -------------------------------------------------------------------------
// Kernel 4: padding mask (bool as 0/1 float).
// -------------------------------------------------------------------------
__global__ __launch_bounds__(256)
void mask_kernel(const int* __restrict__ a_scope, float* __restrict__ out3)
{
    const int idx = blockIdx.x * 256 + threadIdx.x;
    if (idx >= B_MOLS * N_ATOMS) return;
    const int m = idx >> 7;
    const int p = idx & 127;
    out3[idx] = (p < a_scope[2 * m + 1]) ? 0.f : 1.f;
}

extern "C" void kernel_launch(void* const* d_in, const int* in_sizes, int n_in,
                              void* d_out, int out_size, void* d_ws, size_t ws_size,
                              hipStream_t stream) {
    const float* f_atoms     = (const float*)d_in[0];
    const float* f_bonds     = (const float*)d_in[1];
    const float* f_atoms_out = (const float*)d_in[2];
    const float* f_bonds_out = (const float*)d_in[3];
    const int*   b2a         = (const int*)d_in[4];
    const int*   b2revb      = (const int*)d_in[5];
    const int*   a_scope     = (const int*)d_in[6];
    const int*   b_scope     = (const int*)d_in[7];
    // d_in[8] = n_atom scalar (128, compile-time constant here)
    const float* W_atom      = (const float*)d_in[9];
    const float* b_atom      = (const float*)d_in[10];
    const float* W_bond      = (const float*)d_in[11];
    const float* b_bond      = (const float*)d_in[12];

    const int Nb = in_sizes[1] / D_IN;

    float* out1 = (float*)d_out;                                   // [128][128][512]
    float* out2 = out1 + (size_t)N_ATOMS * B_MOLS * D_MODEL;       // [128][16][128][128]
    float* out3 = out2 + (size_t)B_MOLS * NHEAD * N_ATOMS * N_ATOMS; // [128][128]

    // apairs init must precede the scatter (same stream => ordered).
    {
        const size_t total4 = (size_t)B_MOLS * NHEAD * N_ATOMS * N_ATOMS / 4;
        apairs_init_kernel<<<(unsigned)((total4 + 255) / 256), 256, 0, stream>>>(a_scope, out2);
    }
    mask_kernel<<<(B_MOLS * N_ATOMS + 255) / 256, 256, 0, stream>>>(a_scope, out3);
    {
        dim3 grid(4, 8, B_MOLS);
        atoms_emb_kernel<<<grid, 256, 0, stream>>>(f_atoms, f_atoms_out, a_scope,
                                                   W_atom, b_atom, out1);
    }
    {
        const int tiles  = (Nb + 15) / 16;
        const int blocks = (tiles + 7) / 8;
        bonds_scatter_kernel<<<blocks, 256, 0, stream>>>(f_bonds, f_bonds_out,
                                                         b2a, b2revb, b_scope,
                                                         W_bond, b_bond, out2, Nb);
    }
}